// ProcrustEs_62371515072569
// MI455X (gfx1250) — compile-verified
//
#include <hip/hip_runtime.h>
#include <hip/hip_bf16.h>

// ---------------------------------------------------------------------------
// ProcrustEs loss on MI455X (gfx1250, wave32).
//
// Math: for each (r,n): A = tail^T head (16x16, K=512).  T = V U^T from SVD
// of A is exactly the orthogonal polar factor of M = A^T.  Since T is
// orthogonal, ||head T - tail||_F^2 = ||head||^2 + ||tail||^2 - 2 tr(T .* M).
// Polar factor computed by Newton-Schulz: Y <- Y (1.5 I - 0.5 Y^T Y),
// Y0 = M/||M||_F.  All 16x16 matmuls use v_wmma_f32_16x16x4_f32.
//
// Phase 1 stages gathered embedding rows into LDS with the CDNA5 async DMA
// path (global_load_async_to_lds_b128, ASYNCcnt) and double-buffers 2x64KB
// of the 320KB WGP LDS so HBM gather latency overlaps WMMA compute.
// ---------------------------------------------------------------------------

typedef float v2f __attribute__((ext_vector_type(2)));
typedef float v8f __attribute__((ext_vector_type(8)));
typedef __attribute__((address_space(3))) float lds_float_t;

#define TD    2048
#define SUBS  128             // TD / 16
#define RREL  128
#define PAIRS 512
#define NMAT  (RREL * SUBS)   // 16384
#define EPSI  1e-6f
#define NS_ITERS 14
#define NCHUNK (PAIRS / 4)    // 128 K-steps of 4 pairs

// Per-lane async global->LDS 16-byte copy (gfx1250 VGLOBAL opcode 98).
// VDST VGPR = LDS byte address (workgroup-relative), VADDR = 64-bit global.
__device__ __forceinline__ void async_b128(const float* gsrc, float* ldst)
{
  unsigned loff = (unsigned)(unsigned long long)(lds_float_t*)ldst;
  unsigned long long ga = (unsigned long long)gsrc;
  asm volatile("global_load_async_to_lds_b128 %0, %1, off"
               :: "v"(loff), "v"(ga) : "memory");
}
__device__ __forceinline__ void wait_async0()
{
  asm volatile("s_wait_asynccnt 0" ::: "memory");
}

// LDS layout: sm[((buf*2 + isTail)*4 + q) * TD + col]   (2 bufs, 4 pairs)
__device__ __forceinline__ float* sm_slot(float* sm, int buf, int isTail, int q)
{
  return sm + (size_t)(((buf * 2 + isTail) * 4 + q) * TD);
}

// ---------------------------------------------------------------------------
// Phase 1: 1 block per relation, 512 threads = 16 waves, 8 subspaces/wave.
// Each K-step: 4 pairs' head+tail rows (64 KB) streamed to LDS by async DMA,
// then one v_wmma_f32_16x16x4_f32 per subspace.
//
// WMMA f32 16x16x4 operand layout (ISA 7.12.2):
//   A (16x4 MxK): lane l -> M=l%16; VGPR v -> K = 2*(l/16)+v
//   B (4x16 KxN): lane l -> N=l%16; VGPR v -> K = 2*(l/16)+v
//   C/D (16x16):  lane l -> N=l%16; VGPR v -> M = v + 8*(l/16)
// ---------------------------------------------------------------------------
__global__ __launch_bounds__(512) void procrustes_phase1(
    const float* __restrict__ emb, const int* __restrict__ pairs,
    float* __restrict__ wsA, float* __restrict__ wsHH, float* __restrict__ wsTT)
{
  extern __shared__ float sm[];   // 2 * 2 * 4 * TD floats = 128 KB

  const int r    = blockIdx.x;
  const int t    = threadIdx.x;
  const int lane = t & 31;
  const int wave = t >> 5;        // 0..15
  const int m    = lane & 15;
  const int h    = lane >> 4;

  v8f   zero = {};
  v8f   acc[8];
  float hh[8], tt[8];
#pragma unroll
  for (int s = 0; s < 8; ++s) { acc[s] = zero; hh[s] = 0.f; tt[s] = 0.f; }

  // --- async stage of one chunk (4 pairs) into buffer `buf` -----------------
  // 512 threads: per k-step all threads copy float4 #t of one row (uniform
  // row index -> scalarized), 8 steps cover 4 pairs x {head,tail}.
  auto stage_chunk = [&](int it, int buf) {
#pragma unroll
    for (int k = 0; k < 8; ++k) {
      int q = k >> 1, isTail = k & 1;
      int row = pairs[(((r * PAIRS) + it * 4 + q) << 1) + isTail];
      row = __builtin_amdgcn_readfirstlane(row);
      const float* src = emb + (size_t)row * TD + (t << 2);
      float* dst = sm_slot(sm, buf, isTail, q) + (t << 2);
      async_b128(src, dst);
    }
  };

  stage_chunk(0, 0);                              // prologue
  for (int it = 0; it < NCHUNK; ++it) {
    const int cur = it & 1;
    wait_async0();                                // my DMA into `cur` landed
    __syncthreads();                              // all waves' DMA landed;
                                                  // all done reading 1-cur
    if (it + 1 < NCHUNK) stage_chunk(it + 1, 1 - cur);

    const float* tp0 = sm_slot(sm, cur, 1, 2 * h + 0);
    const float* tp1 = sm_slot(sm, cur, 1, 2 * h + 1);
    const float* hp0 = sm_slot(sm, cur, 0, 2 * h + 0);
    const float* hp1 = sm_slot(sm, cur, 0, 2 * h + 1);
#pragma unroll
    for (int s = 0; s < 8; ++s) {
      int idx = (wave * 8 + s) * 16 + m;
      float a0 = tp0[idx], a1 = tp1[idx];         // tail -> A operand
      float b0 = hp0[idx], b1 = hp1[idx];         // head -> B operand
      v2f A = {a0, a1};
      v2f B = {b0, b1};
      acc[s] = __builtin_amdgcn_wmma_f32_16x16x4_f32(
                   false, A, false, B, (short)0, acc[s], false, false);
      tt[s] += a0 * a0 + a1 * a1;
      hh[s] += b0 * b0 + b1 * b1;
    }
  }

  // Write A row-major (A[i][j] at base + i*16 + j) plus reduced hh/tt.
#pragma unroll
  for (int s = 0; s < 8; ++s) {
    int n = wave * 8 + s;
    size_t base = (size_t)(r * SUBS + n) * 256;
#pragma unroll
    for (int v = 0; v < 8; ++v)
      wsA[base + (v + 8 * h) * 16 + m] = acc[s][v];

    float hv = hh[s], tv = tt[s];
#pragma unroll
    for (int o = 16; o >= 1; o >>= 1) {
      hv += __shfl_xor(hv, o, 32);
      tv += __shfl_xor(tv, o, 32);
    }
    if (lane == 0) { wsHH[r * SUBS + n] = hv; wsTT[r * SUBS + n] = tv; }
  }
}

// ---------------------------------------------------------------------------
// Phase 2: Newton-Schulz polar iteration with WMMA; one wave per matrix,
// 8 waves per block, wave-private LDS tiles.
// ---------------------------------------------------------------------------
__global__ __launch_bounds__(256) void procrustes_phase2(
    const float* __restrict__ wsA, const float* __restrict__ wsHH,
    const float* __restrict__ wsTT, float* __restrict__ wsNorm)
{
  __shared__ float Ybuf[8 * 256];
  __shared__ float Zbuf[8 * 256];

  const int t    = threadIdx.x;
  const int lane = t & 31;
  const int wave = t >> 5;
  const int m    = lane & 15;
  const int h    = lane >> 4;
  const int mat  = blockIdx.x * 8 + wave;          // exact grid: no tail

  float* Yw = &Ybuf[wave * 256];
  float* Zw = &Zbuf[wave * 256];
  const float* Asrc = wsA + (size_t)mat * 256;

  // Load A (+ eps*I) in C/D layout: acd[v] = A[v+8h][m].
  float acd[8];
  float fs = 0.f;
#pragma unroll
  for (int v = 0; v < 8; ++v) {
    float a = Asrc[(v + 8 * h) * 16 + m];
    if ((v + 8 * h) == m) a += EPSI;
    acd[v] = a;
    fs += a * a;
  }
#pragma unroll
  for (int o = 16; o >= 1; o >>= 1) fs += __shfl_xor(fs, o, 32);
  const float fnorm = sqrtf(fs);       // ||A||_F  (>= eps, never 0)
  const float inv   = 1.0f / fnorm;

  // Y0 = M / ||M||_F where M = (A+eps I)^T: Yw[row*16+col] = inv*M[row][col].
#pragma unroll
  for (int v = 0; v < 8; ++v) Yw[m * 16 + (v + 8 * h)] = acd[v] * inv;

  // mcd = inv * M in C/D layout (element [v+8h][m]) for the trace later.
  float mcd[8];
#pragma unroll
  for (int v = 0; v < 8; ++v) mcd[v] = Yw[(v + 8 * h) * 16 + m];

  v8f zero = {};
  v8f ycd  = zero;
  for (int itn = 0; itn < NS_ITERS; ++itn) {
    // Z = Y^T Y : Z[i,j] = sum_p Y[p,i] Y[p,j]  (A and B operands identical).
    v8f zacc = zero;
#pragma unroll
    for (int c = 0; c < 4; ++c) {
      float a0 = Yw[(4 * c + 2 * h + 0) * 16 + m];
      float a1 = Yw[(4 * c + 2 * h + 1) * 16 + m];
      v2f A = {a0, a1};
      zacc = __builtin_amdgcn_wmma_f32_16x16x4_f32(
                 false, A, false, A, (short)0, zacc, false, false);
    }
    // Z' = 1.5 I - 0.5 Z, stored row-major.
#pragma unroll
    for (int v = 0; v < 8; ++v) {
      float z = -0.5f * zacc[v];
      if ((v + 8 * h) == m) z += 1.5f;
      Zw[(v + 8 * h) * 16 + m] = z;
    }
    // Ynew = Y * Z'.
    v8f w = zero;
#pragma unroll
    for (int c = 0; c < 4; ++c) {
      float a0 = Yw[m * 16 + (4 * c + 2 * h + 0)];   // Y[p=m][k]
      float a1 = Yw[m * 16 + (4 * c + 2 * h + 1)];
      float b0 = Zw[(4 * c + 2 * h + 0) * 16 + m];   // Z'[k][j=m]
      float b1 = Zw[(4 * c + 2 * h + 1) * 16 + m];
      v2f A = {a0, a1};
      v2f B = {b0, b1};
      w = __builtin_amdgcn_wmma_f32_16x16x4_f32(
              false, A, false, B, (short)0, w, false, false);
    }
    ycd = w;
#pragma unroll
    for (int v = 0; v < 8; ++v) Yw[(v + 8 * h) * 16 + m] = w[v];
  }

  // trace(T .* M) ; mcd holds inv*M so multiply back by fnorm.
  float tr = 0.f;
#pragma unroll
  for (int v = 0; v < 8; ++v) tr += ycd[v] * mcd[v];
#pragma unroll
  for (int o = 16; o >= 1; o >>= 1) tr += __shfl_xor(tr, o, 32);
  tr *= fnorm;

  if (lane == 0) {
    float n2 = wsHH[mat] + wsTT[mat] - 2.0f * tr;
    wsNorm[mat] = sqrtf(fmaxf(n2, 0.0f));
  }
}

// ---------------------------------------------------------------------------
// Phase 3: deterministic fixed-order reduction of the 16384 norms.
// ---------------------------------------------------------------------------
__global__ __launch_bounds__(256) void procrustes_phase3(
    const float* __restrict__ wsNorm, float* __restrict__ out)
{
  __shared__ float red[256];
  const int t = threadIdx.x;
  float s = 0.f;
  for (int i = t; i < NMAT; i += 256) s += wsNorm[i];
  red[t] = s;
  __syncthreads();
  for (int o = 128; o >= 1; o >>= 1) {
    if (t < o) red[t] += red[t + o];
    __syncthreads();
  }
  if (t == 0) out[0] = red[0] * (1.0f / (float)RREL);
}

// ---------------------------------------------------------------------------
extern "C" void kernel_launch(void* const* d_in, const int* in_sizes, int n_in,
                              void* d_out, int out_size, void* d_ws, size_t ws_size,
                              hipStream_t stream)
{
  (void)in_sizes; (void)n_in; (void)out_size; (void)ws_size;
  const float* emb   = (const float*)d_in[0];   // [40000, 2048] f32
  const int*   pairs = (const int*)d_in[1];     // [128, 512, 2] i32
  float*       out   = (float*)d_out;           // scalar

  float* wsA    = (float*)d_ws;                 // NMAT * 256 floats (16 MB)
  float* wsHH   = wsA  + (size_t)NMAT * 256;
  float* wsTT   = wsHH + NMAT;
  float* wsNorm = wsTT + NMAT;

  // Dynamic LDS: 2 buffers * (4 pairs * {head,tail}) * 2048 floats = 128 KB
  // (within the 320 KB CDNA5 WGP LDS).
  const size_t ldsBytes = (size_t)2 * 2 * 4 * TD * sizeof(float);

  procrustes_phase1<<<RREL, 512, ldsBytes, stream>>>(emb, pairs, wsA, wsHH, wsTT);
  procrustes_phase2<<<NMAT / 8, 256, 0, stream>>>(wsA, wsHH, wsTT, wsNorm);
  procrustes_phase3<<<1, 256, 0, stream>>>(wsNorm, out);
}